// SpatialNumDualDescriptorTS3_15848429322647
// MI455X (gfx1250) — compile-verified
//
#include <hip/hip_runtime.h>
#include <math.h>

typedef __attribute__((ext_vector_type(2))) float v2f;
typedef __attribute__((ext_vector_type(8))) float v8f;

#define MDIM 8
#define OB   5
#define T320 (MDIM * MDIM * OB)      // 320 = flattened (i,j,g), t = i*40 + j*5 + g
#define N1   47                      // window starts per axis (48 - 2 + 1)
#define NP   (N1 * N1 * N1)          // 103823 output points
#define NTILES ((NP + 15) / 16)      // 16-point tiles for WMMA (6489)
#define NP_PAD (NTILES * 16)         // 103824: xw padded so stores need no guard
#define TWO_PI 6.28318530717958647692f

// grid strides (floats): [48,48,48,8] row-major
#define SA (48 * 48 * 8)             // 18432
#define SB (48 * 8)                  // 384
#define SC 8

// workspace layout in floats:
//   [0,            3*47*320)        : phi tables (phi1, phi2, phi3), each 47*320
//   [3*47*320, +NP_PAD*8)           : xw (padded to full tiles)
#define WS_XW_OFF (3 * N1 * T320)

// ---------------------------------------------------------------------------
// Kernel 1: phi tables.  phiT[t_table][k][t] = cos(2*pi*k / (t + 2 + t_table))
// grid = (47, 3), block = 320
// ---------------------------------------------------------------------------
__global__ void phi_kernel(float* __restrict__ ws) {
    const int tab = blockIdx.y;          // 0 -> shift 2, 1 -> shift 3, 2 -> shift 4
    const int k   = blockIdx.x;          // 0..46
    const int t   = threadIdx.x;         // 0..319
    const float period = (float)t + 2.0f + (float)tab;
    ws[(size_t)tab * (N1 * T320) + (size_t)k * T320 + t] =
        cosf(TWO_PI * (float)k / period);
}

// ---------------------------------------------------------------------------
// Kernel 2: xw = (1/8) * windowsum(grid) @ M^T  via V_WMMA_F32_16X16X4_F32.
// One wave per 16-point tile; K=8 split into two K=4 WMMAs. N=8 of 16 used.
// A (16x4 f32): lanes 0-15 hold K={0,1} in V0/V1; lanes 16-31 hold K={2,3}.
// B (4x16 f32): lane = N; same half-lane K split.  B[k][n] = Mw[n*8 + k].
// D: VGPR r -> point row r (lanes 0-15) / r+8 (lanes 16-31), N = lane&15.
// Stores are unconditional: xw is padded to NP_PAD rows in the workspace.
// ---------------------------------------------------------------------------
__global__ void xw_kernel(const float* __restrict__ grid,
                          const float* __restrict__ Mw,
                          float* __restrict__ xw) {
    const int lane = threadIdx.x & 31;
    const int wave = blockIdx.x * 8 + (threadIdx.x >> 5);
    if (wave >= NTILES) return;          // wave-uniform: EXEC stays all-1s

    const int half = lane >> 4;          // 0: lanes 0-15, 1: lanes 16-31
    const int l16  = lane & 15;
    const int kb   = half * 2;           // K base within each K=4 WMMA

    // B operands (columns n>=8 unused -> zero)
    const bool nv = (l16 < MDIM);
    v2f B0, B1;
    B0[0] = nv ? Mw[l16 * MDIM + kb + 0] : 0.0f;
    B0[1] = nv ? Mw[l16 * MDIM + kb + 1] : 0.0f;
    B1[0] = nv ? Mw[l16 * MDIM + kb + 4] : 0.0f;
    B1[1] = nv ? Mw[l16 * MDIM + kb + 5] : 0.0f;

    // A operands: window sums of grid for this lane's point row, scaled by 1/8
    const int p  = wave * 16 + l16;
    const int pc = (p < NP) ? p : (NP - 1);      // clamp loads (last tile only)
    const int a  = pc / (N1 * N1);
    const int r0 = pc - a * (N1 * N1);
    const int b  = r0 / N1;
    const int c  = r0 - b * N1;
    const float* g0 = grid + ((size_t)a * SA + (size_t)b * SB + (size_t)c * SC);

    float s[4];                                   // k = kb, kb+1, kb+4, kb+5
#pragma unroll
    for (int q = 0; q < 4; ++q) {
        const int k = kb + (q & 1) + (q >> 1) * 4;
        const float* gk = g0 + k;
        float acc = gk[0]        + gk[SC]
                  + gk[SB]       + gk[SB + SC]
                  + gk[SA]       + gk[SA + SC]
                  + gk[SA + SB]  + gk[SA + SB + SC];
        s[q] = acc * 0.125f;
    }
    v2f A0; A0[0] = s[0]; A0[1] = s[1];
    v2f A1; A1[0] = s[2]; A1[1] = s[3];

    v8f C = {};
    C = __builtin_amdgcn_wmma_f32_16x16x4_f32(false, A0, false, B0,
                                              (short)0, C, false, false);
    C = __builtin_amdgcn_wmma_f32_16x16x4_f32(false, A1, false, B1,
                                              (short)0, C, false, false);

    if (nv) {
        float* dst = xw + ((size_t)(wave * 16 + half * 8)) * MDIM + l16;
#pragma unroll
        for (int r = 0; r < 8; ++r) {
            dst[(size_t)r * MDIM] = C[r];        // padded: no bounds check
        }
    }
}

// ---------------------------------------------------------------------------
// Kernel 3: Nk[a,b,c,i] = sum_{j,g} pp[i,j,g] * phi3[c,i,j,g] * xw[a,b,c,j]
// with pp[t] = phi1[a,t]*phi2[b,t]*P[t] staged once per (a,b) block in LDS.
// grid = (47,47), block = 256
// ---------------------------------------------------------------------------
__global__ void nk_kernel(const float* __restrict__ P,
                          const float* __restrict__ ws,
                          float* __restrict__ out) {
    const int a = blockIdx.x;
    const int b = blockIdx.y;
    const float* phi1 = ws + (size_t)a * T320;
    const float* phi2 = ws + (size_t)(N1 * T320) + (size_t)b * T320;
    const float* phi3 = ws + (size_t)(2 * N1 * T320);
    const float* xw   = ws + WS_XW_OFF;

    __shared__ float pp[T320];
    for (int t = threadIdx.x; t < T320; t += blockDim.x)
        pp[t] = phi1[t] * phi2[t] * P[t];
    __syncthreads();

    for (int o = threadIdx.x; o < N1 * MDIM; o += blockDim.x) {
        const int c = o >> 3;
        const int i = o & 7;
        const size_t pbase = ((size_t)(a * N1 + b) * N1 + c);
        const float* ph3 = phi3 + (size_t)c * T320 + i * (MDIM * OB);
        const float* ppi = pp + i * (MDIM * OB);
        const float* xwp = xw + pbase * MDIM;
        float acc = 0.0f;
#pragma unroll
        for (int j = 0; j < MDIM; ++j) {
            const float xj = xwp[j];
#pragma unroll
            for (int g = 0; g < OB; ++g) {
                const int t = j * OB + g;
                acc = fmaf(ppi[t] * ph3[t], xj, acc);
            }
        }
        out[pbase * MDIM + i] = acc;
    }
}

// ---------------------------------------------------------------------------
extern "C" void kernel_launch(void* const* d_in, const int* in_sizes, int n_in,
                              void* d_out, int out_size, void* d_ws, size_t ws_size,
                              hipStream_t stream) {
    (void)in_sizes; (void)n_in; (void)out_size; (void)ws_size;
    const float* grid = (const float*)d_in[0];   // [48,48,48,8] f32
    const float* Mw   = (const float*)d_in[1];   // [8,8] f32
    const float* P    = (const float*)d_in[2];   // [8,8,5] f32
    float* ws  = (float*)d_ws;
    float* out = (float*)d_out;

    // 1) cosine tables (3 * 47 * 320 floats)
    phi_kernel<<<dim3(N1, 3), 320, 0, stream>>>(ws);

    // 2) xw via f32 WMMA: one wave per 16-point tile, 8 waves per block
    const int blocks = (NTILES + 7) / 8;
    xw_kernel<<<blocks, 256, 0, stream>>>(grid, Mw, ws + WS_XW_OFF);

    // 3) final contraction, one block per (a,b)
    nk_kernel<<<dim3(N1, N1), 256, 0, stream>>>(P, ws, out);
}